// GraphSAGEModule_41412074668542
// MI455X (gfx1250) — compile-verified
//
#include <hip/hip_runtime.h>
#include <hip/hip_bf16.h>

typedef __attribute__((ext_vector_type(2))) float v2f;
typedef __attribute__((ext_vector_type(8))) float v8f;
typedef __attribute__((ext_vector_type(4))) int   v4i;

#define NOUT 64  // every output feature dim is 64

#if __has_builtin(__builtin_amdgcn_global_load_async_to_lds_b128)
#define HAVE_ASYNC_LDS 1
#else
#define HAVE_ASYNC_LDS 0
#endif

#define AS_GLOBAL __attribute__((address_space(1)))
#define AS_LOCAL  __attribute__((address_space(3)))

// ---------------------------------------------------------------------------
// zero fill
// ---------------------------------------------------------------------------
__global__ __launch_bounds__(256)
void zero_f32(float* __restrict__ p, long long n) {
    long long i = (long long)blockIdx.x * blockDim.x + threadIdx.x;
    long long stride = (long long)gridDim.x * blockDim.x;
    for (; i < n; i += stride) p[i] = 0.0f;
}

// ---------------------------------------------------------------------------
// Edge scatter over PRE-PROJECTED (64-wide) features:
//   agg[dst] += p_src[src], cnt[dst] += 1   (only dst < dst_limit)
// One thread per (edge, float4 chunk): 16 threads/edge. Source table (25.6MB)
// and destination buffer (25.6MB) are both L2-resident (192MB), so the
// global_atomic_add_f32 ops resolve in L2. Projecting before scattering
// halves the dominant atomic traffic vs scattering raw 128-wide features.
// ---------------------------------------------------------------------------
__global__ __launch_bounds__(256)
void scatter_mean_accum(const float* __restrict__ xsrc,
                        const int* __restrict__ ei,   // [2, E]: src row then dst row
                        long long E,
                        float* __restrict__ agg, float* __restrict__ cnt,
                        int dst_limit) {
    long long gid = (long long)blockIdx.x * blockDim.x + threadIdx.x;
    long long e = gid >> 4;               // 64/4 = 16 chunks per edge
    int cidx = (int)(gid & 15);
    if (e >= E) return;
    int src = ei[e];
    int dst = ei[E + e];
    if ((unsigned)dst >= (unsigned)dst_limit) return;
    const float4 v = *(const float4*)(xsrc + (long long)src * NOUT + cidx * 4);
    float* ap = agg + (long long)dst * NOUT + cidx * 4;
    atomicAdd(ap + 0, v.x);
    atomicAdd(ap + 1, v.y);
    atomicAdd(ap + 2, v.z);
    atomicAdd(ap + 3, v.w);
    if (cidx == 0) atomicAdd(cnt + dst, 1.0f);
}

// ---------------------------------------------------------------------------
// Unified fp32 WMMA GEMM with fused elementwise epilogue:
//   out[r,c] = act( alpha * ( (X @ W)[r,c]
//                             + (bl   ? bl[c]                        : 0)
//                             + (agg  ? agg[r,c]/max(cnt[r],1)      : 0)
//                             + (prev ? prev[r,c]                   : 0) ) )
// X: M x K (K in {64,128}), W: K x 64, out/agg/prev stride 64.
// Block = 128 threads (4 waves); block owns 16 rows; each wave computes one
// 16x16 tile with a v_wmma_f32_16x16x4_f32 chain. X tile staged in LDS via
// the gfx1250 async global->LDS path (ASYNCcnt) when available.
// ---------------------------------------------------------------------------
__global__ __launch_bounds__(128)
void wmma_gemm_fused(const float* __restrict__ X, const float* __restrict__ W,
                     const float* __restrict__ agg, const float* __restrict__ cnt,
                     const float* __restrict__ bl, const float* __restrict__ prev,
                     float* __restrict__ out,
                     int M, int K, float alpha, int do_relu) {
    const int KP = K + 4;                 // LDS pad (16B units): kills bank conflicts
    __shared__ __align__(16) float sX[16 * (128 + 4)];

    const int tid  = threadIdx.x;
    const int lane = tid & 31;
    const int wave = tid >> 5;
    const long long row0 = (long long)blockIdx.x * 16;

    // ---- stage 16xK X tile into LDS (rows clamped; M is a multiple of 16
    //      for every launch in this pipeline, clamp is safety only) ----
#if HAVE_ASYNC_LDS
    const int kc = K >> 2;                // float4 chunks per row
    for (int i = tid; i < 16 * kc; i += 128) {
        int r  = i / kc;
        int k4 = (i - r * kc) << 2;
        long long gr = row0 + r; if (gr >= M) gr = M - 1;
        __builtin_amdgcn_global_load_async_to_lds_b128(
            (AS_GLOBAL v4i*)(X + gr * K + k4),
            (AS_LOCAL  v4i*)(&sX[r * KP + k4]), 0, 0);
    }
#if __has_builtin(__builtin_amdgcn_s_wait_asynccnt)
    __builtin_amdgcn_s_wait_asynccnt(0);
#else
    asm volatile("s_wait_asynccnt 0" ::: "memory");
#endif
    __syncthreads();
#else
    for (int i = tid; i < 16 * K; i += 128) {
        int r = i / K, k = i - r * K;
        long long gr = row0 + r; if (gr >= M) gr = M - 1;
        sX[r * KP + k] = X[gr * K + k];
    }
    __syncthreads();
#endif

    // Fragment index math per ISA §7.12.2 (wave32):
    //  A (16x4 f32, 2 VGPRs): lanes 0-15 hold K=k+0/k+1, lanes 16-31 K=k+2/k+3
    //  B (4x16 f32, 2 VGPRs): row K striped over lanes, mirrored halves
    //  C/D (16x16 f32, 8 VGPRs): VGPR j -> M=j (lanes 0-15), M=j+8 (lanes 16-31)
    const int n0   = wave * 16;
    const int arow = lane & 15;
    const int kh   = (lane >> 4) << 1;    // 0 or 2
    const int ncol = n0 + (lane & 15);

    v8f c = {};
    for (int k = 0; k < K; k += 4) {
        v2f a, b;
        a[0] = sX[arow * KP + k + kh];
        a[1] = sX[arow * KP + k + kh + 1];
        b[0] = W[(k + kh) * NOUT + ncol];
        b[1] = W[(k + kh + 1) * NOUT + ncol];
        c = __builtin_amdgcn_wmma_f32_16x16x4_f32(false, a, false, b,
                                                  (short)0, c, false, false);
    }

    // ---- fused epilogue ----
    const int rbase = (lane >> 4) * 8;
    const float bias = bl ? bl[ncol] : 0.0f;
    #pragma unroll
    for (int j = 0; j < 8; ++j) {
        long long gr = row0 + rbase + j;
        if (gr >= M) continue;
        float v = c[j] + bias;
        if (agg)  v += agg[gr * NOUT + ncol] * (1.0f / fmaxf(cnt[gr], 1.0f));
        if (prev) v += prev[gr * NOUT + ncol];
        v *= alpha;
        if (do_relu) v = fmaxf(v, 0.0f);
        out[gr * NOUT + ncol] = v;
    }
}

// ---------------------------------------------------------------------------
// host-side orchestration
// ---------------------------------------------------------------------------
extern "C" void kernel_launch(void* const* d_in, const int* in_sizes, int n_in,
                              void* d_out, int out_size, void* d_ws, size_t ws_size,
                              hipStream_t stream) {
    const int F  = 128;   // input feature dim
    const int D  = 64;    // repr dim
    const int BS = 8192;  // batch size (fixed by reference)

    const float* x_user = (const float*)d_in[0];
    const float* x_item = (const float*)d_in[1];
    const int*   ei_ui  = (const int*)d_in[2];
    const int*   ei_iu  = (const int*)d_in[3];
    const int*   ei_uu  = (const int*)d_in[4];
    // d_in[5] = batch_size (device scalar; fixed 8192)
    const float* Wl1_ui = (const float*)d_in[6];
    const float* bl1_ui = (const float*)d_in[7];
    const float* Wr1_ui = (const float*)d_in[8];
    const float* Wl1_iu = (const float*)d_in[9];
    const float* bl1_iu = (const float*)d_in[10];
    const float* Wr1_iu = (const float*)d_in[11];
    const float* Wl1_uu = (const float*)d_in[12];
    const float* bl1_uu = (const float*)d_in[13];
    const float* Wr1_uu = (const float*)d_in[14];
    // layer-2 "ui" params (d_in[15..17]) unused by reference output
    const float* Wl2_iu = (const float*)d_in[18];
    const float* bl2_iu = (const float*)d_in[19];
    const float* Wr2_iu = (const float*)d_in[20];
    const float* Wl2_uu = (const float*)d_in[21];
    const float* bl2_uu = (const float*)d_in[22];
    const float* Wr2_uu = (const float*)d_in[23];

    const long long NU = in_sizes[0] / F;
    const long long NI = in_sizes[1] / F;
    const long long E_ui = (long long)in_sizes[2] / 2;
    const long long E_iu = (long long)in_sizes[3] / 2;
    const long long E_uu = (long long)in_sizes[4] / 2;
    const long long NMAX = (NU > NI) ? NU : NI;

    // workspace layout (floats): ~105 MB total
    float* ws      = (float*)d_ws;
    float* projBuf = ws;                       // NMAX * 64  (projected src feats)
    float* agg64   = projBuf + NMAX * D;       // NMAX * 64  (scatter destination)
    float* cntA    = agg64 + NMAX * D;         // NMAX       (contiguous with agg64)
    float* item1   = cntA + NMAX;              // NI * 64
    float* user1   = item1 + NI * D;           // NU * 64
    float* tmp2    = user1 + NU * D;           // BS * 64

    float* out = (float*)d_out;                // [BS, 64] fp32

    const int ZB = 2048;
    auto sgrid = [](long long work) { return (unsigned)((work + 255) / 256); };
    const unsigned gNU = (unsigned)((NU + 15) / 16);
    const unsigned gNI = (unsigned)((NI + 15) / 16);
    const unsigned gBS = (unsigned)(BS / 16);

    // ================= layer 1, relation ui : item1 =================
    zero_f32<<<ZB, 256, 0, stream>>>(agg64, NMAX * D + NMAX);      // agg64 + cntA
    wmma_gemm_fused<<<gNU, 128, 0, stream>>>(                      // P = x_user @ Wl1_ui
        x_user, Wl1_ui, nullptr, nullptr, nullptr, nullptr, projBuf, (int)NU, F, 1.0f, 0);
    scatter_mean_accum<<<sgrid(E_ui * 16), 256, 0, stream>>>(
        projBuf, ei_ui, E_ui, agg64, cntA, (int)NI);
    wmma_gemm_fused<<<gNI, 128, 0, stream>>>(                      // item1 = relu(agg/cnt + b + x_item@Wr)
        x_item, Wr1_ui, agg64, cntA, bl1_ui, nullptr, item1, (int)NI, F, 1.0f, 1);

    // ================= layer 1, relation iu : user1 (partial) ======
    zero_f32<<<ZB, 256, 0, stream>>>(agg64, NMAX * D + NMAX);
    wmma_gemm_fused<<<gNI, 128, 0, stream>>>(
        x_item, Wl1_iu, nullptr, nullptr, nullptr, nullptr, projBuf, (int)NI, F, 1.0f, 0);
    scatter_mean_accum<<<sgrid(E_iu * 16), 256, 0, stream>>>(
        projBuf, ei_iu, E_iu, agg64, cntA, (int)NU);
    wmma_gemm_fused<<<gNU, 128, 0, stream>>>(
        x_user, Wr1_iu, agg64, cntA, bl1_iu, nullptr, user1, (int)NU, F, 1.0f, 0);

    // ======== layer 1, relation uu : user1 = relu(0.5*(iu+uu)) =====
    zero_f32<<<ZB, 256, 0, stream>>>(agg64, NMAX * D + NMAX);
    wmma_gemm_fused<<<gNU, 128, 0, stream>>>(
        x_user, Wl1_uu, nullptr, nullptr, nullptr, nullptr, projBuf, (int)NU, F, 1.0f, 0);
    scatter_mean_accum<<<sgrid(E_uu * 16), 256, 0, stream>>>(
        projBuf, ei_uu, E_uu, agg64, cntA, (int)NU);
    wmma_gemm_fused<<<gNU, 128, 0, stream>>>(
        x_user, Wr1_uu, agg64, cntA, bl1_uu, user1, user1, (int)NU, F, 0.5f, 1);

    // ================= layer 2 (only dst < BS matters) ==============
    // relation iu
    zero_f32<<<256, 256, 0, stream>>>(agg64, (long long)BS * D);
    zero_f32<<<32, 256, 0, stream>>>(cntA, BS);
    wmma_gemm_fused<<<gNI, 128, 0, stream>>>(                      // P2 = item1 @ Wl2_iu
        item1, Wl2_iu, nullptr, nullptr, nullptr, nullptr, projBuf, (int)NI, D, 1.0f, 0);
    scatter_mean_accum<<<sgrid(E_iu * 16), 256, 0, stream>>>(
        projBuf, ei_iu, E_iu, agg64, cntA, BS);
    wmma_gemm_fused<<<gBS, 128, 0, stream>>>(                      // tmp2 = agg/cnt + b + user1@Wr2_iu
        user1, Wr2_iu, agg64, cntA, bl2_iu, nullptr, tmp2, BS, D, 1.0f, 0);

    // relation uu
    zero_f32<<<256, 256, 0, stream>>>(agg64, (long long)BS * D);
    zero_f32<<<32, 256, 0, stream>>>(cntA, BS);
    wmma_gemm_fused<<<gNU, 128, 0, stream>>>(                      // P2 = user1 @ Wl2_uu
        user1, Wl2_uu, nullptr, nullptr, nullptr, nullptr, projBuf, (int)NU, D, 1.0f, 0);
    scatter_mean_accum<<<sgrid(E_uu * 16), 256, 0, stream>>>(
        projBuf, ei_uu, E_uu, agg64, cntA, BS);
    wmma_gemm_fused<<<gBS, 128, 0, stream>>>(                      // out = 0.5*(tmp2 + agg/cnt + b + user1@Wr2_uu)
        user1, Wr2_uu, agg64, cntA, bl2_uu, tmp2, out, BS, D, 0.5f, 0);
}